// Diffusion_34119220199810
// MI455X (gfx1250) — compile-verified
//
#include <hip/hip_runtime.h>
#include <math.h>

// Problem constants (match reference)
#define V 6144
#define C 16
#define ORDER 25

// Tiling
#define KSPLIT 8
#define KSEG (V / KSPLIT)                       // 768, multiple of 4
#define WAVES_PER_BLOCK 4
#define ROWS_PER_WAVE 32                        // two 16-row WMMA tiles per wave
#define ROWS_PER_BLOCK (WAVES_PER_BLOCK * ROWS_PER_WAVE)  // 128
#define GRID_X (V / ROWS_PER_BLOCK)             // 48

typedef __attribute__((ext_vector_type(2))) float v2f;
typedef __attribute__((ext_vector_type(8))) float v8f;

// ---------------------------------------------------------------------------
// One Taylor step, K-split partial GEMM: part[ksplit] = L[:, kseg] @ term[kseg, :]
// Full f32 path via V_WMMA_F32_16X16X4_F32 (precision matches the f32 reference;
// the op is bandwidth-bound: 25 x 151 MB of L traffic vs 30 GFLOP total, so f32
// WMMA throughput is far beyond what's needed and precision is preserved).
// ---------------------------------------------------------------------------
__global__ __launch_bounds__(WAVES_PER_BLOCK * 32)
void diff_step(const float* __restrict__ Lm,     // V x V row-major
               const float* __restrict__ term,   // V x C row-major
               float* __restrict__ part)         // KSPLIT * V * C partials
{
    const int lane = threadIdx.x & 31;
    const int wave = threadIdx.x >> 5;
    const int half = lane >> 4;    // 0: lanes 0-15, 1: lanes 16-31
    const int l15  = lane & 15;
    const int row0 = blockIdx.x * ROWS_PER_BLOCK + wave * ROWS_PER_WAVE;
    const int k0   = blockIdx.y * KSEG;

    // A fragment (16x4 f32): lane l<16 -> row l, K={k,k+1}; lane 16+l -> K={k+2,k+3}
    const float* A0 = Lm + (size_t)(row0 + l15) * V + (size_t)(2 * half) + k0;
    const float* A1 = A0 + (size_t)16 * V;       // second 16-row tile
    // B fragment (4x16 f32), mirrored K split across lane halves
    const float* B  = term + (size_t)(k0 + 2 * half) * C + l15;

    v8f acc0 = {};
    v8f acc1 = {};

#pragma unroll 4
    for (int k = 0; k < KSEG; k += 4) {
        v2f a0 = *(const v2f*)(A0 + k);          // global_load_b64
        v2f a1 = *(const v2f*)(A1 + k);
        v2f b;
        b.x = B[0];                              // row k+2*half   (coalesced b32)
        b.y = B[C];                              // row k+2*half+1
        B += 4 * C;
        acc0 = __builtin_amdgcn_wmma_f32_16x16x4_f32(
                   false, a0, false, b, (short)0, acc0, false, false);
        acc1 = __builtin_amdgcn_wmma_f32_16x16x4_f32(
                   false, a1, false, b, (short)0, acc1, false, false);
    }

    // C/D layout: VGPR j -> M = j + 8*half, N = l15. Coalesced 64B row stores.
    float* out = part + (size_t)blockIdx.y * ((size_t)V * C);
#pragma unroll
    for (int j = 0; j < 8; ++j) {
        const int r = row0 + j + 8 * half;
        out[(size_t)r * C + l15]        = acc0[j];
        out[(size_t)(r + 16) * C + l15] = acc1[j];
    }
}

// ---------------------------------------------------------------------------
// Deterministic fixed-order reduction of KSPLIT partials + per-channel scale,
// fused with the series accumulation: nxt = s * sum(part); acc += nxt.
// ---------------------------------------------------------------------------
__global__ void diff_reduce(const float* __restrict__ part,
                            const float* __restrict__ t,
                            float* __restrict__ nxt,
                            float* __restrict__ acc,
                            float kinv)
{
    const int i = blockIdx.x * blockDim.x + threadIdx.x;
    if (i >= V * C) return;
    const float s = -fmaxf(t[i & (C - 1)], 1e-8f) * kinv;
    float v = 0.0f;
#pragma unroll
    for (int p = 0; p < KSPLIT; ++p)
        v += part[i + (size_t)p * ((size_t)V * C)];
    v *= s;
    nxt[i] = v;
    acc[i] += v;
}

// term_0 = x ; acc = x   (k=0 Taylor term)
__global__ void diff_init(const float* __restrict__ x,
                          float* __restrict__ term0,
                          float* __restrict__ acc)
{
    const int i = blockIdx.x * blockDim.x + threadIdx.x;
    if (i >= V * C) return;
    const float v = x[i];
    term0[i] = v;
    acc[i]   = v;
}

extern "C" void kernel_launch(void* const* d_in, const int* in_sizes, int n_in,
                              void* d_out, int out_size, void* d_ws, size_t ws_size,
                              hipStream_t stream)
{
    (void)in_sizes; (void)n_in; (void)out_size; (void)ws_size;

    const float* x  = (const float*)d_in[0];   // (V, C)
    const float* Lm = (const float*)d_in[1];   // (V, V)
    const float* t  = (const float*)d_in[2];   // (C,)
    float* acc = (float*)d_out;                // (V, C)

    // Scratch layout: [termA | termB | part0..part<KSPLIT-1>], all V*C floats each.
    float* termA = (float*)d_ws;
    float* termB = termA + (size_t)V * C;
    float* part  = termB + (size_t)V * C;

    const int NE = V * C;
    diff_init<<<(NE + 255) / 256, 256, 0, stream>>>(x, termA, acc);

    float* cur = termA;
    float* nxt = termB;
    for (int k = 1; k <= ORDER; ++k) {
        diff_step<<<dim3(GRID_X, KSPLIT), WAVES_PER_BLOCK * 32, 0, stream>>>(Lm, cur, part);
        diff_reduce<<<(NE + 255) / 256, 256, 0, stream>>>(part, t, nxt, acc,
                                                          1.0f / (float)k);
        float* tmp = cur; cur = nxt; nxt = tmp;
    }
}